// PromptGenerator_65644280152916
// MI455X (gfx1250) — compile-verified
//
#include <hip/hip_runtime.h>
#include <hip/hip_bf16.h>

typedef __attribute__((ext_vector_type(2))) float v2f;
typedef __attribute__((ext_vector_type(4))) float v4f;
typedef __attribute__((ext_vector_type(8))) float v8f;

#define D_DIM   1024
#define N_EXP   128
#define B_SZ    1024
#define PT_D    28
#define TPAD    32
#define KSEL    32   // k-1 = 32 gates kept

// x * v_rcp_f32(1+exp(-x)) : avoids the IEEE v_div_fmas/div_fixup sequence
__device__ __forceinline__ float silu_f(float x) {
    return x * __builtin_amdgcn_rcpf(1.0f + __expf(-x));
}
__device__ __forceinline__ float softplus_f(float x) {
    return (x > 0.0f) ? (x + log1pf(__expf(-x))) : log1pf(__expf(x));
}

// ---------------------------------------------------------------------------
// Kernel A: t_embed table [32 x 1024] = SiLU(t*w1 + b1) @ w2^T + b2
// via V_WMMA_F32_16X16X4_F32. One wave (32 threads) per 16x16 output tile.
// M = padded timestep (2 tiles), N = embed dim j (64 tiles), K = d (256 x 4).
// A-frag (16x4 f32): lanes 0-15 hold K=k0,k0+1 ; lanes 16-31 hold K=k0+2,k0+3.
// B-frag (4x16 f32): lane&15 = column n, same K-half split.
// ---------------------------------------------------------------------------
__global__ void __launch_bounds__(32)
tembed_table_kernel(const float* __restrict__ w1, const float* __restrict__ b1,
                    const float* __restrict__ w2, const float* __restrict__ b2,
                    float* __restrict__ table) {
    const int lane  = threadIdx.x & 31;
    const int mtile = blockIdx.x & 1;
    const int ntile = blockIdx.x >> 1;
    const int khalf = (lane >> 4) << 1;                 // 0 or 2
    const float tval = (float)(mtile * 16 + (lane & 15));
    const int j = ntile * 16 + (lane & 15);
    const float* __restrict__ w2row = w2 + (size_t)j * D_DIM;

    v8f c = {};
    for (int k0 = 0; k0 < D_DIM; k0 += 4) {
        const int kb = k0 + khalf;
        v2f a, b;
        a.x = silu_f(tval * w1[kb]     + b1[kb]);
        a.y = silu_f(tval * w1[kb + 1] + b1[kb + 1]);
        b.x = w2row[kb];
        b.y = w2row[kb + 1];
        c = __builtin_amdgcn_wmma_f32_16x16x4_f32(false, a, false, b,
                                                  (short)0, c, false, false);
    }
    const float bias = b2[j];
#pragma unroll
    for (int r = 0; r < 8; ++r) {
        const int trow = mtile * 16 + r + ((lane >> 4) << 3);  // M = r / r+8
        table[trow * D_DIM + j] = c[r] + bias;
    }
}

// ---------------------------------------------------------------------------
// Kernel B: per (t, n): clean_logit table and noise-std table.
// clean[t,n] = dot(table[t], gate_w[n,:D]) + step*gate_w[n,D] + gate_b[n]
// std  [t,n] = softplus(dot(table[t], w_noise[:,n])) + 0.01
// One wave per (t,n); 28*128 = 3584 waves.
// ---------------------------------------------------------------------------
__global__ void __launch_bounds__(256)
gate_tables_kernel(const float* __restrict__ table,
                   const float* __restrict__ gate_w,
                   const float* __restrict__ gate_b,
                   const float* __restrict__ w_noise,
                   const int*   __restrict__ ts,
                   float* __restrict__ clean_tab,
                   float* __restrict__ std_tab) {
    const int wave = (int)((blockIdx.x * blockDim.x + threadIdx.x) >> 5);
    const int lane = threadIdx.x & 31;
    if (wave >= PT_D * N_EXP) return;
    const int t = wave >> 7;
    const int n = wave & 127;
    const float* __restrict__ trow  = table + (size_t)t * D_DIM;
    const float* __restrict__ gwrow = gate_w + (size_t)n * (D_DIM + 1);
    float s1 = 0.0f, s2 = 0.0f;
    for (int d = lane; d < D_DIM; d += 32) {
        const float h = trow[d];
        s1 += h * gwrow[d];
        s2 += h * w_noise[(size_t)d * N_EXP + n];
    }
#pragma unroll
    for (int off = 16; off > 0; off >>= 1) {
        s1 += __shfl_xor(s1, off);
        s2 += __shfl_xor(s2, off);
    }
    if (lane == 0) {
        const float stepf = (float)ts[0];
        clean_tab[wave] = s1 + stepf * gwrow[D_DIM] + gate_b[n];
        std_tab[wave]   = softplus_f(s2) + 0.01f;
    }
}

// ---------------------------------------------------------------------------
// Kernel C: gather t_embed output rows: out2[b,:] = table[timestep[b],:]
// One block per batch row; ts[b] is block-uniform -> scalar load.
// ---------------------------------------------------------------------------
__global__ void __launch_bounds__(256)
tembed_out_kernel(const float* __restrict__ table, const int* __restrict__ ts,
                  float* __restrict__ out2) {
    const int b  = blockIdx.x;
    const int d4 = threadIdx.x;
    const v4f* __restrict__ src = (const v4f*)(table + (size_t)ts[b] * D_DIM);
    ((v4f*)out2)[(size_t)b * 256 + d4] = src[d4];
}

// ---------------------------------------------------------------------------
// Kernel D: noisy top-32 + softmax -> gates[B, 128]. One wave per row.
// 64-bit ordered keys (monotone float bits | index) + shfl_xor argmax, 32 rounds.
// ---------------------------------------------------------------------------
__global__ void __launch_bounds__(256)
gating_kernel(const float* __restrict__ clean_tab,
              const float* __restrict__ std_tab,
              const float* __restrict__ noise,
              const int*   __restrict__ ts,
              float* __restrict__ gates) {
    const int b    = (int)((blockIdx.x * blockDim.x + threadIdx.x) >> 5);
    const int lane = threadIdx.x & 31;
    if (b >= B_SZ) return;
    const int tb = ts[b];

    unsigned long long key[4];
    float gval[4];
#pragma unroll
    for (int s = 0; s < 4; ++s) {
        const int n = s * 32 + lane;
        const float v = clean_tab[tb * N_EXP + n] +
                        noise[(size_t)b * N_EXP + n] * std_tab[tb * N_EXP + n];
        const unsigned int bits = __float_as_uint(v);
        const unsigned int ord  = (bits & 0x80000000u) ? ~bits : (bits | 0x80000000u);
        key[s]  = ((unsigned long long)ord << 32) | (unsigned int)n;
        gval[s] = 0.0f;
    }

    float m = 0.0f, sum = 0.0f;
    for (int r = 0; r < KSEL; ++r) {
        unsigned long long best = key[0];
#pragma unroll
        for (int s = 1; s < 4; ++s) best = (key[s] > best) ? key[s] : best;
#pragma unroll
        for (int off = 16; off > 0; off >>= 1) {
            const unsigned long long o = __shfl_xor(best, off);
            best = (o > best) ? o : best;
        }
        const unsigned int ord  = (unsigned int)(best >> 32);
        const unsigned int bits = (ord & 0x80000000u) ? (ord & 0x7fffffffu) : ~ord;
        const float fv = __uint_as_float(bits);
        if (r == 0) m = fv;                      // global max, softmax shift
        const float e = __expf(fv - m);
        sum += e;
        const int nbest = (int)(best & 0xffffffffu);
        if (lane == (nbest & 31)) {              // owning lane
            gval[nbest >> 5] = e;
            key[nbest >> 5]  = 0ull;             // remove from candidates
        }
    }
    const float inv = 1.0f / sum;
#pragma unroll
    for (int s = 0; s < 4; ++s)
        gates[(size_t)b * N_EXP + s * 32 + lane] = gval[s] * inv;
}

// ---------------------------------------------------------------------------
// Kernel E: out[b,n,:] = prompts[step,n,:] * gates[b,n]. 512 MB streaming
// store (non-temporal b128). One block = one (b,n) pair, so the gate value
// and prompt-row base are block-uniform (scalar loads); prompts slice
// (512 KB) + gates (512 KB) stay L2-resident across the whole batch.
// ---------------------------------------------------------------------------
__global__ void __launch_bounds__(256)
output_kernel(const float* __restrict__ prompts, const int* __restrict__ ts,
              const float* __restrict__ gates, float* __restrict__ out) {
    const int n  = blockIdx.x & 127;
    const int b  = blockIdx.x >> 7;
    const int d4 = threadIdx.x;
    const int step = ts[0];
    const float g = gates[(size_t)b * N_EXP + n];
    const v4f* __restrict__ p4 =
        (const v4f*)(prompts + ((size_t)step * N_EXP + n) * D_DIM);
    const v4f r = p4[d4] * g;
    __builtin_nontemporal_store(r, ((v4f*)out) + (size_t)blockIdx.x * 256 + d4);
}

// ---------------------------------------------------------------------------
extern "C" void kernel_launch(void* const* d_in, const int* in_sizes, int n_in,
                              void* d_out, int out_size, void* d_ws, size_t ws_size,
                              hipStream_t stream) {
    const float* prompts = (const float*)d_in[0];
    const float* w1      = (const float*)d_in[1];
    const float* b1      = (const float*)d_in[2];
    const float* w2      = (const float*)d_in[3];
    const float* b2      = (const float*)d_in[4];
    const float* gate_w  = (const float*)d_in[5];
    const float* gate_b  = (const float*)d_in[6];
    const float* w_noise = (const float*)d_in[7];
    const float* noise   = (const float*)d_in[8];
    const int*   ts      = (const int*)d_in[9];

    float* ws        = (float*)d_ws;
    float* table     = ws;                                 // 32*1024 floats
    float* clean_tab = table + TPAD * D_DIM;               // 28*128
    float* std_tab   = clean_tab + PT_D * N_EXP;           // 28*128
    float* gates     = std_tab + PT_D * N_EXP;             // 1024*128

    float* out_main   = (float*)d_out;                     // B*N*D
    float* out_tembed = out_main + (size_t)B_SZ * N_EXP * D_DIM; // B*D

    // A: WMMA table GEMM (128 waves, one per 16x16 tile)
    tembed_table_kernel<<<128, 32, 0, stream>>>(w1, b1, w2, b2, table);
    // B: clean-logit / noise-std tables (3584 waves)
    gate_tables_kernel<<<(PT_D * N_EXP + 7) / 8, 256, 0, stream>>>(
        table, gate_w, gate_b, w_noise, ts, clean_tab, std_tab);
    // C: t_embed output gather (4 MB)
    tembed_out_kernel<<<B_SZ, 256, 0, stream>>>(table, ts, out_tembed);
    // D: noisy top-32 softmax gating (1024 waves)
    gating_kernel<<<B_SZ / 8, 256, 0, stream>>>(clean_tab, std_tab, noise, ts, gates);
    // E: 512 MB streaming broadcast-multiply, one block per (b, n)
    output_kernel<<<B_SZ * N_EXP, 256, 0, stream>>>(prompts, ts, gates, out_main);
}